// ExponentialSmoothing_5669356834060
// MI455X (gfx1250) — compile-verified
//
#include <hip/hip_runtime.h>
#include <math.h>

typedef float v2f __attribute__((ext_vector_type(2)));
typedef float v8f __attribute__((ext_vector_type(8)));

#define Bn   16
#define Tn   4096
#define Hn   8
#define Dn   64
#define NCH  16                 // chunks along t
#define CHUNK (Tn / NCH)        // 256 timesteps per chunk
#define TILES (CHUNK / 16)      // 16 WMMA tiles per chunk
#define Sn   (Bn * Hn * Dn)     // 8192 independent series
#define ROW  (Hn * Dn)          // 512 floats = t-stride

// ---------------------------------------------------------------------------
// Kernel 1: per (chunk, series) zero-init local scan endpoint
//   E_c = sum_{i<CHUNK} (1-w) * w^{CHUNK-1-i} * x[c*CHUNK+i]
// Coalesced: consecutive lanes = consecutive d. Pulls input into L2.
// ---------------------------------------------------------------------------
__global__ void k_chunk_tail(const float* __restrict__ x,
                             const float* __restrict__ sw,
                             float* __restrict__ wsE) {
    int idx = blockIdx.x * blockDim.x + threadIdx.x;   // NCH*Sn threads
    int dI = idx & 63;
    int hI = (idx >> 6) & 7;
    int bI = (idx >> 9) & 15;
    int c  = idx >> 13;
    float w   = 1.0f / (1.0f + __expf(-sw[hI]));
    float omw = 1.0f - w;
    const float* p = x + ((size_t)bI * Tn + (size_t)c * CHUNK) * ROW + hI * Dn + dI;
    float e = 0.0f;
#pragma unroll 4
    for (int i = 0; i < CHUNK; ++i)
        e = fmaf(w, e, omw * p[(size_t)i * ROW]);
    wsE[c * Sn + ((bI * Hn + hI) * Dn + dI)] = e;
}

// ---------------------------------------------------------------------------
// Kernel 2: per series, scan the NCH chunk carries (tiny)
//   carry[0] = v0;  carry[c] = w^CHUNK * carry[c-1] + E_{c-1}
// ---------------------------------------------------------------------------
__global__ void k_carry(const float* __restrict__ wsE,
                        const float* __restrict__ sw,
                        const float* __restrict__ v0,
                        float* __restrict__ wsC) {
    int s  = blockIdx.x * blockDim.x + threadIdx.x;    // Sn threads
    int dI = s & 63;
    int hI = (s >> 6) & 7;
    float w  = 1.0f / (1.0f + __expf(-sw[hI]));
    float wL = w;
#pragma unroll
    for (int i = 0; i < 8; ++i) wL = wL * wL;          // w^256
    float carry = v0[hI * Dn + dI];
    wsC[s] = carry;
    for (int c = 1; c < NCH; ++c) {
        carry = fmaf(wL, carry, wsE[(c - 1) * Sn + s]);
        wsC[c * Sn + s] = carry;
    }
}

// ---------------------------------------------------------------------------
// Kernel 3: main scan. One wave32 per (chunk, b, h, d-block of 16).
// Per 16x16 tile:  D = M(16x16) @ X(16x16) + diag(w^{i+1}) (x) carry
// via 4 chained v_wmma_f32_16x16x4_f32. Carry crosses tiles by shuffling
// row 15 (VGPR7 of lanes 16..31).
// ---------------------------------------------------------------------------
__global__ void k_scan(const float* __restrict__ x,
                       const float* __restrict__ sw,
                       const float* __restrict__ wsC,
                       float* __restrict__ out) {
    int wid  = blockIdx.x * (blockDim.x >> 5) + (threadIdx.x >> 5);
    int lane = threadIdx.x & 31;
    int half = lane >> 4;
    int n    = lane & 15;          // tile column (d) for B/C/D; row for A

    int dblk = wid & 3;
    int hI   = (wid >> 2) & 7;
    int bI   = (wid >> 5) & 15;
    int c    = wid >> 9;           // chunk

    float w   = 1.0f / (1.0f + __expf(-sw[hI]));
    float omw = 1.0f - w;

    float wpow[17];
    wpow[0] = 1.0f;
    for (int j = 1; j <= 16; ++j) wpow[j] = wpow[j - 1] * w;

    // A-matrix slices: 16x4 f32 layout — lane = row r, VGPR0/1 hold K = (2*half, 2*half+1)
    int r = n;
    v2f a[4];
#pragma unroll
    for (int k = 0; k < 4; ++k) {
        int c0 = 4 * k + 2 * half;
        a[k].x = (r >= c0)     ? omw * wpow[r - c0]     : 0.0f;
        a[k].y = (r >= c0 + 1) ? omw * wpow[r - c0 - 1] : 0.0f;
    }
    // C-build factors: row M = j + 8*half needs w^{M+1}
    float cm[8];
#pragma unroll
    for (int j = 0; j < 8; ++j) cm[j] = wpow[j + 8 * half + 1];

    int scol   = (bI * Hn + hI) * Dn + dblk * 16 + n;
    float carry = wsC[c * Sn + scol];

    const float* px = x   + ((size_t)bI * Tn + (size_t)c * CHUNK) * ROW + hI * Dn + dblk * 16 + n;
    float*       po = out + ((size_t)bI * Tn + (size_t)c * CHUNK) * ROW + hI * Dn + dblk * 16 + n;

    for (int tile = 0; tile < TILES; ++tile) {
        const float* pt = px + (size_t)tile * 16 * ROW;
        if (tile + 1 < TILES) {                         // global_prefetch_b8 next tile
            __builtin_prefetch(pt + (size_t)16 * ROW, 0, 3);
            __builtin_prefetch(pt + (size_t)24 * ROW, 0, 3);
        }
        // B tile: 4x16 f32 layout — lane = col n, VGPR0/1 hold K rows (2*half, 2*half+1)
        v2f bm[4];
#pragma unroll
        for (int k = 0; k < 4; ++k) {
            int row = 4 * k + 2 * half;
            bm[k].x = pt[(size_t)row * ROW];
            bm[k].y = pt[(size_t)(row + 1) * ROW];
        }
        v8f acc;
#pragma unroll
        for (int j = 0; j < 8; ++j) acc[j] = cm[j] * carry;

        acc = __builtin_amdgcn_wmma_f32_16x16x4_f32(false, a[0], false, bm[0], (short)0, acc, false, false);
        acc = __builtin_amdgcn_wmma_f32_16x16x4_f32(false, a[1], false, bm[1], (short)0, acc, false, false);
        acc = __builtin_amdgcn_wmma_f32_16x16x4_f32(false, a[2], false, bm[2], (short)0, acc, false, false);
        acc = __builtin_amdgcn_wmma_f32_16x16x4_f32(false, a[3], false, bm[3], (short)0, acc, false, false);

        float* qo = po + (size_t)tile * 16 * ROW;
#pragma unroll
        for (int j = 0; j < 8; ++j)
            qo[(size_t)(j + 8 * half) * ROW] = acc[j];

        // next-tile carry: row 15 lives in acc[7] of lanes 16..31 (col = lane-16)
        carry = __shfl(acc[7], 16 + n, 32);
    }
}

// ---------------------------------------------------------------------------
extern "C" void kernel_launch(void* const* d_in, const int* in_sizes, int n_in,
                              void* d_out, int out_size, void* d_ws, size_t ws_size,
                              hipStream_t stream) {
    (void)in_sizes; (void)n_in; (void)out_size; (void)ws_size;
    const float* values = (const float*)d_in[0];   // [16,4096,8,64]
    const float* sw     = (const float*)d_in[1];   // [8,1]
    const float* v0     = (const float*)d_in[2];   // [1,1,8,64]
    float* out = (float*)d_out;
    float* wsE = (float*)d_ws;                     // [NCH][Sn]
    float* wsC = wsE + NCH * Sn;                   // [NCH][Sn]  (1 MB total)

    k_chunk_tail<<<(NCH * Sn) / 256, 256, 0, stream>>>(values, sw, wsE);
    k_carry<<<Sn / 256, 256, 0, stream>>>(wsE, sw, v0, wsC);
    // 8192 full waves, 8 waves per 256-thread block -> 1024 blocks, no partial waves
    k_scan<<<(NCH * Bn * Hn * (Dn / 16)) / 8, 256, 0, stream>>>(values, sw, wsC, out);
}